// DendSN_42872363549253
// MI455X (gfx1250) — compile-verified
//
#include <hip/hip_runtime.h>

// Problem constants (from reference): T=8, N=16, C=128, H=W=32
// B=4 compartments, K=C/B=32, L=H*W=1024
// sites = N*K*L = 524288, processed 16 per wave.
#define T_STRIDE_X   2097152   // N*C*L elements between time steps of x
#define N_STRIDE_X    131072   // C*L
#define T_STRIDE_OUT  524288   // N*K*L elements between time steps of out
#define TOTAL_SITES   524288
#define SITES_PER_WAVE 16

typedef __attribute__((ext_vector_type(2))) float v2f;
typedef __attribute__((ext_vector_type(8))) float v8f;

// A[m][s] = (s<=m<8) ? 0.5^(m-s) : 0   (alpha = 0.5 -> 2^(s-m))
__device__ __forceinline__ float aval(int m, int s) {
    return (m < 8 && s <= m) ? __builtin_exp2f((float)(s - m)) : 0.0f;
}

__global__ __launch_bounds__(256) void
dendsn_wmma_kernel(const float* __restrict__ x,
                   const float* __restrict__ kparam,
                   float* __restrict__ out) {
    const int lane  = threadIdx.x & 31;
    const int wave  = blockIdx.x * (blockDim.x >> 5) + (threadIdx.x >> 5);
    const int site  = wave * SITES_PER_WAVE + (lane & 15);

    // site -> (n, kk, l): site = n*(32*1024) + kk*1024 + l
    const int n  = site >> 15;
    const int kk = (site >> 10) & 31;
    const int l  = site & 1023;
    // element offset of x[t=0, n, channel=kk*4+b=0, l]
    const int siteBase = n * N_STRIDE_X + kk * 4096 + l;

    // half-wave split supplies K-rows {0,1} (lanes 0-15) or {2,3} (lanes 16-31)
    const int thalf = (lane >> 4) << 1;   // 0 or 2
    const int m     = lane & 15;          // M row (= time t) held by this lane

    // A-matrix fragments for the two K=4 steps (k0 = 0 and 4), constant per lane
    v2f a0, a1;
    a0.x = aval(m, thalf + 0);
    a0.y = aval(m, thalf + 1);
    a1.x = aval(m, thalf + 4);
    a1.y = aval(m, thalf + 5);

    // softmax over the 4 dendritic strengths, fold in the *3 gain
    const float k0v = kparam[0], k1v = kparam[1], k2v = kparam[2], k3v = kparam[3];
    const float mx  = fmaxf(fmaxf(k0v, k1v), fmaxf(k2v, k3v));
    const float e0 = __expf(k0v - mx), e1 = __expf(k1v - mx),
                e2 = __expf(k2v - mx), e3 = __expf(k3v - mx);
    const float inv = 3.0f / (e0 + e1 + e2 + e3);
    const float fs[4] = { e0 * inv, e1 * inv, e2 * inv, e3 * inv };

    // mexican hat constants: v = 0.75 + 1e-5
    const float VARV = 0.75f + 1e-5f;
    const float INVV = 1.0f / VARV;
    const float C1   = rsqrtf(6.283185307179586f * VARV);  // 1/sqrt(2*pi*v)

    float y[8];
#pragma unroll
    for (int t = 0; t < 8; ++t) y[t] = 0.0f;

#pragma unroll
    for (int b = 0; b < 4; ++b) {
        const int off = siteBase + b * 1024;
        // B-matrix fragments: lane = column (site), V0/V1 = K-rows thalf(+k0), +1
        v2f bv0, bv1;
        bv0.x = x[(thalf + 0) * T_STRIDE_X + off];
        bv0.y = x[(thalf + 1) * T_STRIDE_X + off];
        bv1.x = x[(thalf + 4) * T_STRIDE_X + off];
        bv1.y = x[(thalf + 5) * T_STRIDE_X + off];

        // pre_act[t][site] = sum_{s=0..7} A[t][s] * x[s][site], exact f32 WMMA
        v8f d = {0.f, 0.f, 0.f, 0.f, 0.f, 0.f, 0.f, 0.f};
        d = __builtin_amdgcn_wmma_f32_16x16x4_f32(false, a0, false, bv0,
                                                  (short)0, d, false, false);
        d = __builtin_amdgcn_wmma_f32_16x16x4_f32(false, a1, false, bv1,
                                                  (short)0, d, false, false);

        const float w = fs[b];
#pragma unroll
        for (int t = 0; t < 8; ++t) {
            const float px = d[t];                // pre_act[t] for this lane's site
            const float u  = px * px * INVV;
            const float g  = C1 * __expf(-0.5f * u);
            y[t] += w * (1.0f - u) * g * INVV;    // fs*3*mexican_hat
        }
    }

    // LIF soma recurrence with hard reset (beta=0.5, v_th=1, v_reset=0)
    float sp[8];
    float h = 0.0f;
#pragma unroll
    for (int t = 0; t < 8; ++t) {
        h = 0.5f * h + y[t];
        const bool fire = (h >= 1.0f);
        sp[t] = fire ? 1.0f : 0.0f;
        h = fire ? 0.0f : h;
    }

    // only lanes 0-15 carry valid D rows (t = 0..7); 16 lanes write one
    // contiguous 64B segment per (t) -> no wasted HBM traffic
    if (lane < 16) {
#pragma unroll
        for (int t = 0; t < 8; ++t) {
            out[t * T_STRIDE_OUT + site] = sp[t];
        }
    }
}

extern "C" void kernel_launch(void* const* d_in, const int* in_sizes, int n_in,
                              void* d_out, int out_size, void* d_ws, size_t ws_size,
                              hipStream_t stream) {
    const float* x  = (const float*)d_in[0];   // (8,16,128,32,32) f32
    const float* kp = (const float*)d_in[1];   // (1,4,1) f32
    float* out = (float*)d_out;                // (8,16,32,32,32) f32

    // 32768 waves * 16 sites = 524288 sites; 256 threads = 8 waves per block
    dim3 grid(TOTAL_SITES / SITES_PER_WAVE / 8);
    dim3 block(256);
    hipLaunchKernelGGL(dendsn_wmma_kernel, grid, block, 0, stream, x, kp, out);
    (void)in_sizes; (void)n_in; (void)out_size; (void)d_ws; (void)ws_size;
}